// MultiSimilarityLoss1_68702296866887
// MI455X (gfx1250) — compile-verified
//
#include <hip/hip_runtime.h>
#include <math.h>

typedef __attribute__((ext_vector_type(2))) float v2f;
typedef __attribute__((ext_vector_type(8))) float v8f;

#define THRESH    0.5f
#define SCALE_POS 2.0f
#define SCALE_NEG 40.0f

#define WMMA_F32(acc, x, y) \
    acc = __builtin_amdgcn_wmma_f32_16x16x4_f32(false, (x), false, (y), (short)0, (acc), false, false)

// One K-step (K += 4) worth of fragments: two 16x4 A strips + four 4x16 B tiles.
struct Frag {
    v2f a0, a1, b0, b1, b2, b3;
};

// All offsets are ints so that with compile-time D they fold into the signed
// 24-bit IOFFSET immediate of global_load_b64 (no per-iteration address VALU).
template <int BS16>
__device__ __forceinline__ Frag load_frag(const float* __restrict__ aP,
                                          const float* __restrict__ bP,
                                          int k) {
    Frag f;
    f.a0 = *(const v2f*)(aP + k);
    f.a1 = *(const v2f*)(aP + BS16 + k);
    f.b0 = *(const v2f*)(bP + k);
    f.b1 = *(const v2f*)(bP + BS16 + k);
    f.b2 = *(const v2f*)(bP + 2 * BS16 + k);
    f.b3 = *(const v2f*)(bP + 3 * BS16 + k);
    return f;
}

#define WMMA8()                          \
    do {                                 \
        WMMA_F32(c00, f0.a0, f0.b0);     \
        WMMA_F32(c01, f0.a0, f0.b1);     \
        WMMA_F32(c02, f0.a0, f0.b2);     \
        WMMA_F32(c03, f0.a0, f0.b3);     \
        WMMA_F32(c10, f0.a1, f0.b0);     \
        WMMA_F32(c11, f0.a1, f0.b1);     \
        WMMA_F32(c12, f0.a1, f0.b2);     \
        WMMA_F32(c13, f0.a1, f0.b3);     \
    } while (0)

// Fused tile kernel: 128x64 output tile per block, 4 waves (wave32).
// Each wave computes a 32x64 strip as 8 independent 16x16 f32-WMMA
// accumulators (2 row strips x 4 col tiles), software-pipelined with
// prefetch distance 2 so ~12 loads are in flight while WMMAs execute.
// DC: compile-time D (0 = dynamic fallback).
template <int DC>
__global__ __launch_bounds__(128) void ms_tile_kernel(
    const float* __restrict__ A,      // [N, D] row-major (emb_img)
    const float* __restrict__ B,      // [M, D] row-major (emb_text)
    const int*   __restrict__ labA,   // [N] gt_pre
    const int*   __restrict__ labB,   // [M] gt_pres_map
    int dRT,
    float*    __restrict__ psum,      // [nblocks]
    float*    __restrict__ nsum,      // [nblocks]
    unsigned* __restrict__ pcnt,      // [nblocks]
    unsigned* __restrict__ ncnt)      // [nblocks]
{
    __shared__ int      rowLab[128];
    __shared__ int      colLab[64];
    __shared__ float    redF[2][4];
    __shared__ unsigned redU[2][4];

    const int D = DC ? DC : dRT;                 // constant-folds for DC=256

    const int tid  = threadIdx.x;
    const int lane = tid & 31;
    const int wave = tid >> 5;                   // 0..3
    const int rowBase = blockIdx.y * 128;
    const int colBase = blockIdx.x * 64;

    rowLab[tid] = labA[rowBase + tid];
    if (tid < 64) colLab[tid] = labB[colBase + tid];
    __syncthreads();

    // f32 16x4 A-fragment / 4x16 B-fragment per-lane addressing:
    // lane l: m = l&15, k-subpair base kb = 2*(l>>4); element pair contiguous.
    const int mr = lane & 15;
    const int kb = (lane >> 4) << 1;

    const float* aP = A + (size_t)(rowBase + wave * 32 + mr) * (size_t)D + kb;
    const float* bP = B + (size_t)(colBase + mr) * (size_t)D + kb;
    constexpr int BS16C = DC ? 16 * DC : 0;      // 16-row stride in floats
    const int BS16 = DC ? BS16C : 16 * dRT;

    v8f c00 = {}, c01 = {}, c02 = {}, c03 = {};
    v8f c10 = {}, c11 = {}, c12 = {}, c13 = {};

    if constexpr (DC != 0) {
        // Fully immediate-offset addressed, prefetch-distance-2 pipeline.
        Frag f0 = load_frag<BS16C>(aP, bP, 0);
        Frag f1 = load_frag<BS16C>(aP, bP, 4);
        #pragma unroll 2
        for (int k = 8; k < DC; k += 4) {
            Frag f2 = load_frag<BS16C>(aP, bP, k);
            WMMA8();
            f0 = f1;
            f1 = f2;
        }
        WMMA8();
        f0 = f1;
        WMMA8();
    } else {
        Frag f0 = load_frag<0>(aP, bP, 0);       // BS16 template 0 unused; use dyn below
        // dynamic-stride fallback (rare path)
        f0.a1 = *(const v2f*)(aP + BS16);
        f0.b1 = *(const v2f*)(bP + BS16);
        f0.b2 = *(const v2f*)(bP + 2 * BS16);
        f0.b3 = *(const v2f*)(bP + 3 * BS16);
        for (int k = 4; k < D; k += 4) {
            Frag f1;
            f1.a0 = *(const v2f*)(aP + k);
            f1.a1 = *(const v2f*)(aP + BS16 + k);
            f1.b0 = *(const v2f*)(bP + k);
            f1.b1 = *(const v2f*)(bP + BS16 + k);
            f1.b2 = *(const v2f*)(bP + 2 * BS16 + k);
            f1.b3 = *(const v2f*)(bP + 3 * BS16 + k);
            WMMA8();
            f0 = f1;
        }
        WMMA8();
    }

    // Epilogue: C/D layout => lane l, vgpr r holds element (M = r + 8*(l>>4), N = l&15)
    float    ps = 0.0f, ns = 0.0f;
    unsigned pc = 0u,   nc = 0u;
    const int rbase = wave * 32 + ((lane >> 4) << 3);  // + strip*16 + r
    const int nloc  = lane & 15;

    auto consume = [&](const v8f& c, int strip, int t) {
        const int cl = colLab[nloc + t * 16];
        const int rb = rbase + strip * 16;
        #pragma unroll
        for (int r = 0; r < 8; ++r) {
            float s = c[r];
            if (s > 0.0f) {
                float d = s - THRESH;
                if (rowLab[rb + r] == cl) { pc++; ps += __expf(-SCALE_POS * d); }
                else                      { nc++; ns += __expf( SCALE_NEG * d); }
            }
        }
    };
    consume(c00, 0, 0); consume(c01, 0, 1); consume(c02, 0, 2); consume(c03, 0, 3);
    consume(c10, 1, 0); consume(c11, 1, 1); consume(c12, 1, 2); consume(c13, 1, 3);

    // wave32 shuffle reduction (fixed order -> deterministic)
    #pragma unroll
    for (int off = 16; off > 0; off >>= 1) {
        ps += __shfl_down(ps, off, 32);
        ns += __shfl_down(ns, off, 32);
        pc += __shfl_down(pc, off, 32);
        nc += __shfl_down(nc, off, 32);
    }
    if (lane == 0) {
        redF[0][wave] = ps; redF[1][wave] = ns;
        redU[0][wave] = pc; redU[1][wave] = nc;
    }
    __syncthreads();
    if (tid == 0) {
        const int bid = blockIdx.y * gridDim.x + blockIdx.x;
        psum[bid] = redF[0][0] + redF[0][1] + redF[0][2] + redF[0][3];
        nsum[bid] = redF[1][0] + redF[1][1] + redF[1][2] + redF[1][3];
        pcnt[bid] = redU[0][0] + redU[0][1] + redU[0][2] + redU[0][3];
        ncnt[bid] = redU[1][0] + redU[1][1] + redU[1][2] + redU[1][3];
    }
}

// Single-block fixed-order final reduction + loss formula.
__global__ __launch_bounds__(256) void ms_reduce_kernel(
    const float* __restrict__ psum, const float* __restrict__ nsum,
    const unsigned* __restrict__ pcnt, const unsigned* __restrict__ ncnt,
    int nblocks, float* __restrict__ out)
{
    __shared__ float    sf[2][256];
    __shared__ unsigned su[2][256];
    const int tid = threadIdx.x;

    float    ps = 0.0f, ns = 0.0f;
    unsigned pc = 0u,   nc = 0u;
    for (int i = tid; i < nblocks; i += 256) {
        ps += psum[i]; ns += nsum[i];
        pc += pcnt[i]; nc += ncnt[i];
    }
    sf[0][tid] = ps; sf[1][tid] = ns;
    su[0][tid] = pc; su[1][tid] = nc;
    __syncthreads();

    for (int s = 128; s > 0; s >>= 1) {
        if (tid < s) {
            sf[0][tid] += sf[0][tid + s];
            sf[1][tid] += sf[1][tid + s];
            su[0][tid] += su[0][tid + s];
            su[1][tid] += su[1][tid + s];
        }
        __syncthreads();
    }

    if (tid == 0) {
        float    PS = sf[0][0], NS = sf[1][0];
        unsigned PC = su[0][0], NC = su[1][0];
        float pos_loss = (PC > 0u) ? (log1pf(PS) / (SCALE_POS * (float)PC)) : 0.0f;
        float neg_loss = (NC > 0u) ? (log1pf(NS) / (SCALE_NEG * (float)NC)) : 0.0f;
        out[0] = pos_loss + neg_loss;
    }
}

extern "C" void kernel_launch(void* const* d_in, const int* in_sizes, int n_in,
                              void* d_out, int out_size, void* d_ws, size_t ws_size,
                              hipStream_t stream) {
    const float* A    = (const float*)d_in[0];   // emb_img  [N, D]
    const float* B    = (const float*)d_in[1];   // emb_text [M, D]
    const int*   labA = (const int*)d_in[2];     // gt_pre   [N]
    const int*   labB = (const int*)d_in[3];     // gt_pres_map [M]

    const int N = in_sizes[2];
    const int M = in_sizes[3];
    const int D = in_sizes[0] / N;               // 256

    const int gx = M / 64;                       // 128
    const int gy = N / 128;                      // 64
    const int nblocks = gx * gy;

    float*    psum = (float*)d_ws;
    float*    nsum = psum + nblocks;
    unsigned* pcnt = (unsigned*)(nsum + nblocks);
    unsigned* ncnt = pcnt + nblocks;

    dim3 grid(gx, gy);
    if (D == 256) {
        ms_tile_kernel<256><<<grid, 128, 0, stream>>>(A, B, labA, labB, D,
                                                      psum, nsum, pcnt, ncnt);
    } else {
        ms_tile_kernel<0><<<grid, 128, 0, stream>>>(A, B, labA, labB, D,
                                                    psum, nsum, pcnt, ncnt);
    }
    ms_reduce_kernel<<<1, 256, 0, stream>>>(psum, nsum, pcnt, ncnt, nblocks,
                                            (float*)d_out);
}